// PIA_88940182766029
// MI455X (gfx1250) — compile-verified
//
#include <hip/hip_runtime.h>
#include <math.h>

#define HH 384
#define WW 384
#define HWPIX (HH*WW)          // 147456
#define BATCH 2
#define CDIM 64
#define QKVC 192
#define HEADS 8
#define CH 8

typedef __attribute__((ext_vector_type(2))) float v2f;
typedef __attribute__((ext_vector_type(8))) float v8f;

// ---- workspace layout (floats) ----
// qkv0: QKVC*HW (per-batch scratch, reused)        28,311,552
// qkv : BATCH*QKVC*HW                              56,623,104
// gram: BATCH*HEADS*64                                  1,024
// qn2 : BATCH*HEADS*8                                     128
// kn2 : BATCH*HEADS*8                                     128
// peff: BATCH*64*64                                     8,192
#define WS_QKV0 0L
#define WS_QKV  ((long)QKVC*HWPIX)
#define WS_GRAM (WS_QKV + (long)BATCH*QKVC*HWPIX)
#define WS_QN2  (WS_GRAM + BATCH*HEADS*CH*CH)
#define WS_KN2  (WS_QN2 + BATCH*HEADS*CH)
#define WS_PEFF (WS_KN2 + BATCH*HEADS*CH)

// -------------------------------------------------------------------------
// Generic 1x1-conv GEMM via V_WMMA_F32_16X16X4_F32.
// Y[b][o][n] = sum_k W[b][o][k] * X[b][k][n],  K fixed = 64.
// One wave computes a 16(out)x16(pix) tile, iterating 8 pixel tiles (128 px)
// to amortize the A (weight) fragment loads. 16 WMMA ops cover K=64.
// -------------------------------------------------------------------------
__global__ __launch_bounds__(256) void conv1x1_wmma(
    const float* __restrict__ X, long xBS,
    const float* __restrict__ W, long wBS,
    float* __restrict__ Y, long yBS,
    int numOT)
{
  const int lane = threadIdx.x & 31;
  const int wave = threadIdx.x >> 5;
  const int b    = blockIdx.z;
  const int gw   = blockIdx.x * 8 + wave;
  const int oT   = gw % numOT;
  const int pg   = gw / numOT;

  const float* Xb = X + (long)b * xBS;
  const float* Wb = W + (long)b * wBS;
  float*       Yb = Y + (long)b * yBS;

  const int m  = lane & 15;    // A row / B,D column
  const int hi = lane >> 4;    // half-wave select

  // Preload A fragments for the whole K=64 loop (16x4 f32 layout:
  // lanes 0-15 hold K = k+0,k+1 ; lanes 16-31 hold K = k+2,k+3).
  v2f a[16];
#pragma unroll
  for (int kk = 0; kk < 16; ++kk) {
    const float* wp = Wb + (long)(oT * 16 + m) * 64 + kk * 4 + 2 * hi;
    a[kk][0] = wp[0];
    a[kk][1] = wp[1];
  }

  const int n0 = pg * 128;
#pragma unroll
  for (int t = 0; t < 8; ++t) {
    const int n = n0 + t * 16 + m;
    v8f acc = {};
#pragma unroll
    for (int kk = 0; kk < 16; ++kk) {
      const int krow = kk * 4 + 2 * hi;   // B 4x16: rows mirrored like A
      v2f bb;
      bb[0] = Xb[(long)krow * HWPIX + n];
      bb[1] = Xb[(long)(krow + 1) * HWPIX + n];
      acc = __builtin_amdgcn_wmma_f32_16x16x4_f32(
          false, a[kk], false, bb, (short)0, acc, false, false);
    }
    // D layout: VGPR r -> row r (lanes 0-15) / row r+8 (lanes 16-31), col = lane&15
#pragma unroll
    for (int r = 0; r < 8; ++r) {
      Yb[(long)(oT * 16 + r + 8 * hi) * HWPIX + n] = acc[r];
    }
  }
}

// -------------------------------------------------------------------------
// Depthwise 3x3 conv, SAME zero padding. One block = 16x16 pixel tile of one
// channel; 18x18 halo staged through LDS.
// -------------------------------------------------------------------------
__global__ __launch_bounds__(256) void dwconv3x3_k(
    const float* __restrict__ in, const float* __restrict__ w9,
    float* __restrict__ out)
{
  __shared__ float tile[18 * 18];
  const int c  = blockIdx.z;               // channel 0..191
  const int bx = blockIdx.x * 16;
  const int by = blockIdx.y * 16;
  const float* inc = in + (long)c * HWPIX;
  const int tid = threadIdx.x;

  for (int i = tid; i < 18 * 18; i += 256) {
    const int ly = i / 18, lx = i - ly * 18;
    const int gy = by + ly - 1, gx = bx + lx - 1;
    float v = 0.f;
    if (gy >= 0 && gy < HH && gx >= 0 && gx < WW) v = inc[gy * WW + gx];
    tile[i] = v;
  }
  __syncthreads();

  const float* wc = w9 + c * 9;
  const int ty = tid >> 4, tx = tid & 15;
  float s = 0.f;
#pragma unroll
  for (int dy = 0; dy < 3; ++dy)
#pragma unroll
    for (int dx = 0; dx < 3; ++dx)
      s += tile[(ty + dy) * 18 + (tx + dx)] * wc[dy * 3 + dx];

  out[(long)c * HWPIX + (by + ty) * WW + (bx + tx)] = s;
}

__device__ __forceinline__ float waveReduce(float v) {
#pragma unroll
  for (int off = 16; off > 0; off >>= 1) v += __shfl_down(v, off, 32);
  return v;
}

// -------------------------------------------------------------------------
// Per (b,head): 8x8 Gram matrix  G[c][d] = sum_n q_c[n]*k_d[n]  plus the
// squared L2 norms of each q/k row. Wave-reduce then atomicAdd partials.
// -------------------------------------------------------------------------
__global__ __launch_bounds__(256) void gram_k(
    const float* __restrict__ qkv, float* __restrict__ gram,
    float* __restrict__ qn2, float* __restrict__ kn2)
{
  const int h = blockIdx.y, b = blockIdx.z;
  const float* qb = qkv + ((long)b * QKVC + h * CH) * HWPIX;
  const float* kb = qkv + ((long)b * QKVC + 64 + h * CH) * HWPIX;

  float g[64], qs[8], ks[8];
#pragma unroll
  for (int i = 0; i < 64; ++i) g[i] = 0.f;
#pragma unroll
  for (int i = 0; i < 8; ++i) { qs[i] = 0.f; ks[i] = 0.f; }

  for (long n = blockIdx.x * 256 + threadIdx.x; n < HWPIX;
       n += (long)gridDim.x * 256) {
    float qv[8], kv[8];
#pragma unroll
    for (int c = 0; c < 8; ++c) {
      qv[c] = qb[(long)c * HWPIX + n];
      kv[c] = kb[(long)c * HWPIX + n];
    }
#pragma unroll
    for (int c = 0; c < 8; ++c) {
      qs[c] += qv[c] * qv[c];
      ks[c] += kv[c] * kv[c];
#pragma unroll
      for (int d = 0; d < 8; ++d) g[c * 8 + d] += qv[c] * kv[d];
    }
  }

  const int lane = threadIdx.x & 31;
  const int base = b * HEADS + h;
#pragma unroll
  for (int i = 0; i < 64; ++i) {
    float v = waveReduce(g[i]);
    if (lane == 0) atomicAdd(&gram[base * 64 + i], v);
  }
#pragma unroll
  for (int c = 0; c < 8; ++c) {
    float vq = waveReduce(qs[c]);
    float vk = waveReduce(ks[c]);
    if (lane == 0) {
      atomicAdd(&qn2[base * 8 + c], vq);
      atomicAdd(&kn2[base * 8 + c], vk);
    }
  }
}

// -------------------------------------------------------------------------
// Tiny 1-block kernel: normalize Gram -> attn, apply temperature, build the
// combined top-k softmax matrix W_eff = sum_i sc_i * softmax_i(masked attn),
// then fold in the projection:  P_eff[b][o][h*8+d] = sum_c proj[o][h*8+c]*W_eff
// -------------------------------------------------------------------------
__global__ __launch_bounds__(256) void combine_k(
    const float* __restrict__ gram, const float* __restrict__ qn2,
    const float* __restrict__ kn2, const float* __restrict__ temperature,
    const float* __restrict__ attn_sc, const float* __restrict__ proj_w,
    float* __restrict__ peff)
{
  __shared__ float weff[BATCH * HEADS * CH * CH];   // 1024 floats
  const int tid = threadIdx.x;

  if (tid < BATCH * HEADS * CH) {               // one thread per attn row
    const int c  = tid & 7;
    const int bh = tid >> 3;                    // b*8+h
    const int h  = bh & 7;
    const float temp = temperature[h];
    const float qn = fmaxf(sqrtf(qn2[bh * 8 + c]), 1e-12f);

    float row[8];
#pragma unroll
    for (int d = 0; d < 8; ++d) {
      const float kn = fmaxf(sqrtf(kn2[bh * 8 + d]), 1e-12f);
      row[d] = gram[bh * 64 + c * 8 + d] / (qn * kn) * temp;
    }
    // sort a copy descending (8 elements)
    float srt[8];
#pragma unroll
    for (int d = 0; d < 8; ++d) srt[d] = row[d];
    for (int i = 1; i < 8; ++i) {
      float key = srt[i];
      int j = i - 1;
      while (j >= 0 && srt[j] < key) { srt[j + 1] = srt[j]; --j; }
      srt[j + 1] = key;
    }
    const int kks[4] = {CH / 2, (2 * CH) / 3, (3 * CH) / 4, (4 * CH) / 5}; // 4,5,6,6
    const float mx = srt[0];
    float acc[8];
#pragma unroll
    for (int d = 0; d < 8; ++d) acc[d] = 0.f;
    for (int i = 0; i < 4; ++i) {
      const float thr = srt[kks[i] - 1];
      float sum = 0.f;
#pragma unroll
      for (int d = 0; d < 8; ++d)
        if (row[d] >= thr) sum += expf(row[d] - mx);
      const float sc = attn_sc[i] / sum;
#pragma unroll
      for (int d = 0; d < 8; ++d)
        if (row[d] >= thr) acc[d] += sc * expf(row[d] - mx);
    }
#pragma unroll
    for (int d = 0; d < 8; ++d) weff[bh * 64 + c * 8 + d] = acc[d];
  }
  __syncthreads();

  for (int idx = tid; idx < BATCH * 64 * 64; idx += 256) {
    const int j = idx & 63;            // input channel h*8+d
    const int o = (idx >> 6) & 63;     // output channel
    const int b = idx >> 12;
    const int h = j >> 3, d = j & 7;
    float s = 0.f;
#pragma unroll
    for (int c = 0; c < 8; ++c)
      s += proj_w[o * 64 + h * 8 + c] * weff[(b * 8 + h) * 64 + c * 8 + d];
    peff[idx] = s;
  }
}

__global__ void zero_k(float* __restrict__ p, int n) {
  const int i = blockIdx.x * 256 + threadIdx.x;
  if (i < n) p[i] = 0.f;
}

extern "C" void kernel_launch(void* const* d_in, const int* in_sizes, int n_in,
                              void* d_out, int out_size, void* d_ws, size_t ws_size,
                              hipStream_t stream) {
  const float* x           = (const float*)d_in[0];
  const float* qkv_w       = (const float*)d_in[1];
  const float* qkv_dw_w    = (const float*)d_in[2];
  const float* temperature = (const float*)d_in[3];
  const float* attn_sc     = (const float*)d_in[4];
  const float* proj_w      = (const float*)d_in[5];
  float* out = (float*)d_out;
  float* ws  = (float*)d_ws;

  float* qkv0 = ws + WS_QKV0;
  float* qkv  = ws + WS_QKV;
  float* gram = ws + WS_GRAM;
  float* qn2  = ws + WS_QN2;
  float* kn2  = ws + WS_KN2;
  float* peff = ws + WS_PEFF;

  // zero gram + qn2 + kn2 (contiguous: 1024+128+128 floats)
  zero_k<<<(1280 + 255) / 256, 256, 0, stream>>>(gram, 1280);

  // Stage 1+2 per batch: qkv0 = qkv_w @ x ; qkv[b] = dwconv3x3(qkv0)
  for (int b = 0; b < BATCH; ++b) {
    const int numOT = QKVC / 16;                  // 12
    const int waves = numOT * (HWPIX / 128);      // 13824
    conv1x1_wmma<<<dim3(waves / 8, 1, 1), 256, 0, stream>>>(
        x + (long)b * CDIM * HWPIX, 0L, qkv_w, 0L, qkv0, 0L, numOT);
    dwconv3x3_k<<<dim3(WW / 16, HH / 16, QKVC), 256, 0, stream>>>(
        qkv0, qkv_dw_w, qkv + (long)b * QKVC * HWPIX);
  }

  // Stage 3: Gram + norms
  gram_k<<<dim3(64, HEADS, BATCH), 256, 0, stream>>>(qkv, gram, qn2, kn2);

  // Stage 4: top-k softmax combine folded with projection -> P_eff
  combine_k<<<1, 256, 0, stream>>>(gram, qn2, kn2, temperature, attn_sc,
                                   proj_w, peff);

  // Stage 5: out[b] = P_eff[b] @ v[b]
  {
    const int numOT = CDIM / 16;                  // 4
    const int waves = numOT * (HWPIX / 128);      // 4608
    conv1x1_wmma<<<dim3(waves / 8, 1, BATCH), 256, 0, stream>>>(
        qkv + 128L * HWPIX, (long)QKVC * HWPIX,   // v plane, batch stride 192*HW
        peff, 64L * 64,
        out, (long)CDIM * HWPIX, numOT);
  }
}